// CenterLoss_19653770346497
// MI455X (gfx1250) — compile-verified
//
#include <hip/hip_runtime.h>

#define B_ROWS 16384
#define C_CLS  10000
#define D_DIM  256

typedef float v2f __attribute__((ext_vector_type(2)));
typedef float v4f __attribute__((ext_vector_type(4)));
typedef float v8f __attribute__((ext_vector_type(8)));

// -------------------------------------------------------------------------
// Kernel 1: per-row cross-entropy NLL. One 256-thread block per row.
// Single memory pass: cache the row in registers (40 floats/thread),
// block-reduce max, then sum of exp2 (native v_exp_f32), write NLL.
// outputs row = 10000 floats = exactly 2500 float4 chunks.
// Threads handle chunks t + 256*j; j<9 always valid (2303 < 2500),
// j==9 valid for t < 196. Non-temporal loads: data is streamed once.
// -------------------------------------------------------------------------
__global__ __launch_bounds__(256) void ce_rows_kernel(
    const float* __restrict__ outputs,
    const int*   __restrict__ target,
    float*       __restrict__ nll_out)
{
    const int row = blockIdx.x;
    const int t   = threadIdx.x;
    const float* rp  = outputs + (size_t)row * C_CLS;
    const v4f*   rp4 = (const v4f*)rp;

    v4f   regs[10];
    float m = -3.4e38f;

    #pragma unroll
    for (int j = 0; j < 9; ++j) {
        v4f v = __builtin_nontemporal_load(&rp4[t + 256 * j]);
        regs[j] = v;
        m = fmaxf(m, fmaxf(fmaxf(v.x, v.y), fmaxf(v.z, v.w)));
    }
    const bool has10 = (t + 2304) < 2500;
    if (has10) {
        v4f v = __builtin_nontemporal_load(&rp4[t + 2304]);
        regs[9] = v;
        m = fmaxf(m, fmaxf(fmaxf(v.x, v.y), fmaxf(v.z, v.w)));
    }

    __shared__ float red[256];
    red[t] = m;
    __syncthreads();
    #pragma unroll
    for (int s = 128; s > 0; s >>= 1) {
        if (t < s) red[t] = fmaxf(red[t], red[t + s]);
        __syncthreads();
    }
    const float M = red[0];
    __syncthreads();

    const float L2E = 1.4426950408889634f;   // log2(e)
    float s = 0.0f;
    #pragma unroll
    for (int j = 0; j < 9; ++j) {
        v4f v = regs[j];
        s += exp2f((v.x - M) * L2E);
        s += exp2f((v.y - M) * L2E);
        s += exp2f((v.z - M) * L2E);
        s += exp2f((v.w - M) * L2E);
    }
    if (has10) {
        v4f v = regs[9];
        s += exp2f((v.x - M) * L2E);
        s += exp2f((v.y - M) * L2E);
        s += exp2f((v.z - M) * L2E);
        s += exp2f((v.w - M) * L2E);
    }

    red[t] = s;
    __syncthreads();
    #pragma unroll
    for (int st = 128; st > 0; st >>= 1) {
        if (t < st) red[t] += red[t + st];
        __syncthreads();
    }
    if (t == 0) {
        const int   tg = target[row];
        const float xt = rp[tg];
        // log(sum e^{x-M}) + M - x_t ; use native base-2 log
        nll_out[row] = 0.6931471805599453f * log2f(red[0]) + M - xt;
    }
}

// -------------------------------------------------------------------------
// Kernel 2: center loss via V_WMMA_F32_16X16X4_F32.
// One wave (32 lanes) handles a tile of 16 rows. Per K-step of 4:
//   A (16x4 of embeddings):  lane L<16 -> (row L,   k+0..1)
//                            lane L>=16 -> (row L-16, k+2..3)
//   B (4x16 gathered centers, col n = center of target[R+n]): same pattern.
// D = A x B + C accumulates; diag(D)[m] = e_{R+m} . c_{t_{R+m}}.
// ||e||^2 / ||c||^2 accumulate as VALU in the same loop; lane pair (L, L+16)
// covers the full K range for row (L&15), combined with shfl_xor(16).
// Diagonal extraction per ISA C/D layout:
//   m<8 : VGPR m   @ lane m        ;  m>=8 : VGPR m-8 @ lane m+16
// -------------------------------------------------------------------------
__global__ __launch_bounds__(256) void center_rows_kernel(
    const float* __restrict__ emb,
    const float* __restrict__ centers,
    const int*   __restrict__ target,
    float*       __restrict__ dist_out)
{
    const int lane = threadIdx.x & 31;
    const int wave = threadIdx.x >> 5;
    const int tile = blockIdx.x * 8 + wave;       // 8 waves per block
    const int R    = tile * 16;                   // first row of tile

    const int r     = lane & 15;                  // row within tile (A) / col (B)
    const int khalf = (lane >> 4) * 2;            // k sub-offset: 0 or 2
    const int rowA  = R + r;
    const int ctr   = target[rowA];

    const float* ep = emb     + (size_t)rowA * D_DIM + khalf;
    const float* cp = centers + (size_t)ctr  * D_DIM + khalf;

    v8f   acc = {};
    float esq = 0.0f, csq = 0.0f;

    #pragma unroll 4
    for (int k = 0; k < D_DIM; k += 4) {
        v2f a = *(const v2f*)(ep + k);
        v2f b = *(const v2f*)(cp + k);
        esq += a.x * a.x + a.y * a.y;
        csq += b.x * b.x + b.y * b.y;
        acc = __builtin_amdgcn_wmma_f32_16x16x4_f32(
            /*neg_a=*/false, a, /*neg_b=*/false, b,
            /*c_mod=*/(short)0, acc, /*reuse_a=*/false, /*reuse_b=*/false);
    }

    // combine the two K-halves of the squared norms for row (lane&15)
    esq += __shfl_xor(esq, 16, 32);
    csq += __shfl_xor(csq, 16, 32);

    // diagonal extraction: lane L picks component (L&7); lanes 0..15 read
    // from src = L (m<8) or L+16 (m>=8), whose pick is exactly acc[diag].
    const int c = lane & 7;
    float pick = acc[0];
    if (c == 1) pick = acc[1];
    if (c == 2) pick = acc[2];
    if (c == 3) pick = acc[3];
    if (c == 4) pick = acc[4];
    if (c == 5) pick = acc[5];
    if (c == 6) pick = acc[6];
    if (c == 7) pick = acc[7];
    const int src = (lane < 8) ? lane : ((lane < 16) ? (lane + 16) : lane);
    const float cross = __shfl(pick, src, 32);

    if (lane < 16) {
        float dist = esq + csq - 2.0f * cross;
        dist = fminf(fmaxf(dist, 1e-12f), 1e12f);
        dist_out[rowA] = dist;
    }
}

// -------------------------------------------------------------------------
// Kernel 3: deterministic single-block reduction of 2*B scratch values.
// result = (sum(nll) + COEF * sum(dist)) / B, with COEF = 1.
// -------------------------------------------------------------------------
__global__ __launch_bounds__(256) void final_reduce_kernel(
    const float* __restrict__ ws,
    float*       __restrict__ out)
{
    const int t = threadIdx.x;
    float s = 0.0f;
    for (int i = t; i < 2 * B_ROWS; i += 256)
        s += ws[i];
    __shared__ float red[256];
    red[t] = s;
    __syncthreads();
    #pragma unroll
    for (int st = 128; st > 0; st >>= 1) {
        if (t < st) red[t] += red[t + st];
        __syncthreads();
    }
    if (t == 0)
        out[0] = red[0] * (1.0f / (float)B_ROWS);
}

// -------------------------------------------------------------------------
extern "C" void kernel_launch(void* const* d_in, const int* in_sizes, int n_in,
                              void* d_out, int out_size, void* d_ws, size_t ws_size,
                              hipStream_t stream)
{
    const float* embeddings = (const float*)d_in[0];   // [B, D]
    const float* outputs    = (const float*)d_in[1];   // [B, C]
    const int*   target     = (const int*)  d_in[2];   // [B]
    const float* centers    = (const float*)d_in[3];   // [C, D]
    float* ws  = (float*)d_ws;                         // [2*B] scratch
    float* out = (float*)d_out;                        // [1]

    // nll per row -> ws[0..B), dist per row -> ws[B..2B)
    ce_rows_kernel<<<B_ROWS, 256, 0, stream>>>(outputs, target, ws);
    center_rows_kernel<<<B_ROWS / (16 * 8), 256, 0, stream>>>(
        embeddings, centers, target, ws + B_ROWS);
    final_reduce_kernel<<<1, 256, 0, stream>>>(ws, out);
}